// GCN_34849364639898
// MI455X (gfx1250) — compile-verified
//
#include <hip/hip_runtime.h>
#include <math.h>

// ---------------------------------------------------------------------------
// GCN autoencoder forward on MI455X (gfx1250).
// fp32 end-to-end: V_WMMA_F32_16X16X4_F32 for every GEMM; Tensor Data Mover
// (tensor_load_to_lds, TENSORcnt-tracked) double-buffered against WMMA
// compute; guarded scalar staging only on ragged edge/tail tiles.
// ---------------------------------------------------------------------------

#define N_NODES 11616
#define HID     512
#define NLAYER  4
#define NEDGE   371712
#define SZ0     4278
#define SZ1     2081
#define SZ2     5257
#define FT0     3066
#define FT1     2081
#define FT2     5257

typedef __attribute__((ext_vector_type(2))) float    v2f;
typedef __attribute__((ext_vector_type(8))) float    v8f;
typedef __attribute__((ext_vector_type(4))) unsigned v4u;
typedef __attribute__((ext_vector_type(8))) unsigned v8u;

static __device__ __forceinline__ v8f wmma_f32(v2f a, v2f b, v8f c) {
  return __builtin_amdgcn_wmma_f32_16x16x4_f32(
      false, a, false, b, (short)0, c, false, false);
}

// Low 32 bits of a flat shared pointer == byte offset within wave LDS.
static __device__ __forceinline__ unsigned lds_off(const void* p) {
  return (unsigned)(uintptr_t)p;
}

// ---------------------------------------------------------------------------
// TDM 2D tile load: tile1 rows x tile0 f32 elements, row stride `stride0`
// elements, into LDS at `ldsByte` with 1 DWORD pad after every 2^(padCode+1)
// DWORDs (padCode: 3 -> per 16 DW, 5 -> per 64 DW). One wave issues; EXEC is
// ignored by TENSOR ops; completion tracked by TENSORcnt (in-order per wave).
// ---------------------------------------------------------------------------
static __device__ __forceinline__ void tdm_load_2d(
    unsigned ldsByte, const float* g, unsigned tile0, unsigned tile1,
    unsigned long long stride0, unsigned padCode)
{
  unsigned long long ga = (unsigned long long)(uintptr_t)g;
  v4u g0;
  g0[0] = 1u;                                   // count=1 (valid user D#)
  g0[1] = ldsByte;                              // lds_addr
  g0[2] = (unsigned)ga;                         // global_addr[31:0]
  g0[3] = (unsigned)((ga >> 32) & 0x01FFFFFFu)  // global_addr[56:32]
        | (2u << 30);                           // type=2 ("image")
  v8u g1;
  g1[0] = (2u << 16)                            // data_size: 4 bytes
        | (1u << 20) | (padCode << 22);         // pad_enable, pad_amount=1 DW
  g1[1] = (tile0 & 0xFFFFu) << 16;              // tensor_dim0 = tile0
  g1[2] = (tile0 >> 16) | ((tile1 & 0xFFFFu) << 16);   // tensor_dim1 = tile1
  g1[3] = (tile1 >> 16) | (tile0 << 16);        // tile_dim0
  g1[4] = tile1;                                // tile_dim1 (tile_dim2 = 0)
  g1[5] = (unsigned)(stride0 & 0xFFFFFFFFull);  // tensor_dim0_stride[31:0]
  g1[6] = (unsigned)((stride0 >> 32) & 0xFFFFull);
  g1[7] = 0u;
  asm volatile("tensor_load_to_lds %0, %1" :: "s"(g0), "s"(g1) : "memory");
}

// Guarded scalar staging (edge / tail tiles only)
static __device__ __forceinline__ void stage_A_guarded(
    float (*sA)[17], const float* __restrict__ A,
    int tileM, int k0, int M, int K, int tid)
{
  const int ar = tid >> 1, ac0 = (tid & 1) << 3;
#pragma unroll
  for (int j = 0; j < 8; ++j) {
    int gr = tileM + ar, gc = k0 + ac0 + j;
    float v = 0.0f;
    if (gr < M && gc < K) v = A[(size_t)gr * K + gc];
    sA[ar][ac0 + j] = v;
  }
}
static __device__ __forceinline__ void stage_B_guarded(
    float (*sB)[65], const float* __restrict__ B,
    int tileN, int k0, int K, int P, int tid)
{
  const int br = tid >> 4, bc0 = (tid & 15) << 2;
#pragma unroll
  for (int j = 0; j < 4; ++j) {
    int gr = k0 + br, gc = tileN + bc0 + j;
    float v = 0.0f;
    if (gr < K && gc < P) v = B[(size_t)gr * P + gc];
    sB[br][bc0 + j] = v;
  }
}

// ---------------------------------------------------------------------------
// Generic fp32 WMMA GEMM: C[M x P] = A[M x K] @ B[K x P] (+ bias[P])
// Block tile 128x64, K-step 16. 8 waves each own a 32x32 quadrant
// (2x2 WMMA tiles). Interior: double-buffered TDM overlapped with WMMA.
// ---------------------------------------------------------------------------
template <int ADD_BIAS>
__global__ __launch_bounds__(256) void gemm_wmma_f32(
    const float* __restrict__ A, const float* __restrict__ B,
    const float* __restrict__ bias, float* __restrict__ C,
    int M, int K, int P)
{
  __shared__ float sA[2][128][17];   // TDM pads 16 DW rows -> stride 17
  __shared__ float sB[2][16][65];    // TDM pads 64 DW rows -> stride 65

  const int tid  = threadIdx.x;
  const int lane = tid & 31;
  const int w    = tid >> 5;
  const int tileM = blockIdx.y * 128;
  const int tileN = blockIdx.x * 64;
  const int wm0 = (w & 3) << 5, wm1 = wm0 + 16;
  const int wn0 = (w >> 2) << 5, wn1 = wn0 + 16;
  const int mi = lane & 15;
  const int kb = (lane >> 4) << 1;

  v8f acc00 = {}, acc01 = {}, acc10 = {}, acc11 = {};

  auto compute = [&](float (*a)[17], float (*b)[65]) {
#pragma unroll
    for (int kk = 0; kk < 16; kk += 4) {
      v2f a0, a1, b0, b1;
      a0.x = a[wm0 + mi][kk + kb];   a0.y = a[wm0 + mi][kk + kb + 1];
      a1.x = a[wm1 + mi][kk + kb];   a1.y = a[wm1 + mi][kk + kb + 1];
      b0.x = b[kk + kb][wn0 + mi];   b0.y = b[kk + kb + 1][wn0 + mi];
      b1.x = b[kk + kb][wn1 + mi];   b1.y = b[kk + kb + 1][wn1 + mi];
      acc00 = wmma_f32(a0, b0, acc00);
      acc01 = wmma_f32(a0, b1, acc01);
      acc10 = wmma_f32(a1, b0, acc10);
      acc11 = wmma_f32(a1, b1, acc11);
    }
  };

  const bool fast = (tileM + 128 <= M) && (tileN + 64 <= P);
  if (fast) {
    const int nfull = K >> 4;
    if (tid < 32) {   // wave 0 drives the Tensor Data Mover
      tdm_load_2d(lds_off(&sA[0][0][0]), A + (size_t)tileM * K,
                  16, 128, (unsigned long long)K, 3);
      tdm_load_2d(lds_off(&sB[0][0][0]), B + tileN,
                  64, 16, (unsigned long long)P, 5);
    }
    for (int s = 0; s < nfull; ++s) {
      const int buf = s & 1;
      if (tid < 32) {
        if (s + 1 < nfull) {   // prefetch next stage into the other buffer
          const int k1 = (s + 1) << 4;
          tdm_load_2d(lds_off(&sA[buf ^ 1][0][0]),
                      A + (size_t)tileM * K + k1, 16, 128,
                      (unsigned long long)K, 3);
          tdm_load_2d(lds_off(&sB[buf ^ 1][0][0]),
                      B + (size_t)k1 * P + tileN, 64, 16,
                      (unsigned long long)P, 5);
          __builtin_amdgcn_s_wait_tensorcnt(2);  // stage s landed
        } else {
          __builtin_amdgcn_s_wait_tensorcnt(0);
        }
      }
      __syncthreads();
      compute(sA[buf], sB[buf]);
      __syncthreads();
    }
    if (K & 15) {   // guarded K tail
      const int k0 = nfull << 4, buf = nfull & 1;
      stage_A_guarded(sA[buf], A, tileM, k0, M, K, tid);
      stage_B_guarded(sB[buf], B, tileN, k0, K, P, tid);
      __syncthreads();
      compute(sA[buf], sB[buf]);
      __syncthreads();
    }
  } else {          // ragged M/N edges: fully guarded single buffer
    for (int k0 = 0; k0 < K; k0 += 16) {
      stage_A_guarded(sA[0], A, tileM, k0, M, K, tid);
      stage_B_guarded(sB[0], B, tileN, k0, K, P, tid);
      __syncthreads();
      compute(sA[0], sB[0]);
      __syncthreads();
    }
  }

  const int rbase = (lane >> 4) << 3;
#pragma unroll
  for (int r = 0; r < 8; ++r) {
    int row0 = tileM + wm0 + rbase + r;
    int row1 = tileM + wm1 + rbase + r;
    int c0 = tileN + wn0 + mi, c1 = tileN + wn1 + mi;
    float bb0 = (ADD_BIAS && c0 < P) ? bias[c0] : 0.0f;
    float bb1 = (ADD_BIAS && c1 < P) ? bias[c1] : 0.0f;
    if (row0 < M) {
      if (c0 < P) C[(size_t)row0 * P + c0] = acc00[r] + bb0;
      if (c1 < P) C[(size_t)row0 * P + c1] = acc01[r] + bb1;
    }
    if (row1 < M) {
      if (c0 < P) C[(size_t)row1 * P + c0] = acc10[r] + bb0;
      if (c1 < P) C[(size_t)row1 * P + c1] = acc11[r] + bb1;
    }
  }
}

// ---------------------------------------------------------------------------
// Fused z = x @ x^T tile + embedding loss: sum( softplus(z) - adj*z ).
// 128x64 z-tile; both operand tiles (rows of x) double-buffered via TDM.
// ---------------------------------------------------------------------------
__global__ __launch_bounds__(256) void zloss_wmma(
    const float* __restrict__ X, const float* __restrict__ adj,
    float* __restrict__ accum)
{
  __shared__ float sXa[2][128][17];
  __shared__ float sXb[2][64][17];
  __shared__ float red[256];

  const int tid  = threadIdx.x;
  const int lane = tid & 31;
  const int w    = tid >> 5;
  const int tileM = blockIdx.y * 128;
  const int tileN = blockIdx.x * 64;
  const int wm0 = (w & 3) << 5, wm1 = wm0 + 16;
  const int wn0 = (w >> 2) << 5, wn1 = wn0 + 16;
  const int mi = lane & 15;
  const int kb = (lane >> 4) << 1;

  v8f acc00 = {}, acc01 = {}, acc10 = {}, acc11 = {};

  auto compute = [&](float (*a)[17], float (*b)[17]) {
#pragma unroll
    for (int kk = 0; kk < 16; kk += 4) {
      v2f a0, a1, b0, b1;
      a0.x = a[wm0 + mi][kk + kb];  a0.y = a[wm0 + mi][kk + kb + 1];
      a1.x = a[wm1 + mi][kk + kb];  a1.y = a[wm1 + mi][kk + kb + 1];
      b0.x = b[wn0 + mi][kk + kb];  b0.y = b[wn0 + mi][kk + kb + 1];
      b1.x = b[wn1 + mi][kk + kb];  b1.y = b[wn1 + mi][kk + kb + 1];
      acc00 = wmma_f32(a0, b0, acc00);
      acc01 = wmma_f32(a0, b1, acc01);
      acc10 = wmma_f32(a1, b0, acc10);
      acc11 = wmma_f32(a1, b1, acc11);
    }
  };

  const bool fast = (tileM + 128 <= N_NODES) && (tileN + 64 <= N_NODES);
  constexpr int NST = HID >> 4;     // 32 stages, K always full
  if (fast) {
    if (tid < 32) {
      tdm_load_2d(lds_off(&sXa[0][0][0]), X + (size_t)tileM * HID,
                  16, 128, (unsigned long long)HID, 3);
      tdm_load_2d(lds_off(&sXb[0][0][0]), X + (size_t)tileN * HID,
                  16, 64, (unsigned long long)HID, 3);
    }
    for (int s = 0; s < NST; ++s) {
      const int buf = s & 1;
      if (tid < 32) {
        if (s + 1 < NST) {
          const int k1 = (s + 1) << 4;
          tdm_load_2d(lds_off(&sXa[buf ^ 1][0][0]),
                      X + (size_t)tileM * HID + k1, 16, 128,
                      (unsigned long long)HID, 3);
          tdm_load_2d(lds_off(&sXb[buf ^ 1][0][0]),
                      X + (size_t)tileN * HID + k1, 16, 64,
                      (unsigned long long)HID, 3);
          __builtin_amdgcn_s_wait_tensorcnt(2);
        } else {
          __builtin_amdgcn_s_wait_tensorcnt(0);
        }
      }
      __syncthreads();
      compute(sXa[buf], sXb[buf]);
      __syncthreads();
    }
  } else {
    for (int s = 0; s < NST; ++s) {
      const int k0 = s << 4;
      {
        const int ar = tid >> 1, ac0 = (tid & 1) << 3;
#pragma unroll
        for (int j = 0; j < 8; ++j) {
          int gr = tileM + ar;
          sXa[0][ar][ac0 + j] =
              (gr < N_NODES) ? X[(size_t)gr * HID + k0 + ac0 + j] : 0.0f;
        }
      }
      {
        const int br = tid >> 2, bc0 = (tid & 3) << 2;
#pragma unroll
        for (int j = 0; j < 4; ++j) {
          int gr = tileN + br;
          sXb[0][br][bc0 + j] =
              (gr < N_NODES) ? X[(size_t)gr * HID + k0 + bc0 + j] : 0.0f;
        }
      }
      __syncthreads();
      compute(sXa[0], sXb[0]);
      __syncthreads();
    }
  }

  float lsum = 0.0f;
  const int rbase = (lane >> 4) << 3;
#pragma unroll
  for (int r = 0; r < 8; ++r) {
    int row0 = tileM + wm0 + rbase + r;
    int row1 = tileM + wm1 + rbase + r;
    int c0 = tileN + wn0 + mi, c1 = tileN + wn1 + mi;
#pragma unroll
    for (int t = 0; t < 4; ++t) {
      int row = (t < 2) ? row0 : row1;
      int col = (t & 1) ? c1 : c0;
      float z = (t == 0) ? acc00[r] : (t == 1) ? acc01[r]
              : (t == 2) ? acc10[r] : acc11[r];
      if (row < N_NODES && col < N_NODES) {
        float sp = fmaxf(z, 0.0f) + log1pf(expf(-fabsf(z)));
        lsum += sp - adj[(size_t)row * N_NODES + col] * z;
      }
    }
  }
  red[tid] = lsum;
  __syncthreads();
  for (int s = 128; s > 0; s >>= 1) {
    if (tid < s) red[tid] += red[tid + s];
    __syncthreads();
  }
  if (tid == 0) atomicAdd(accum, red[0]);
}

// ---------------------------------------------------------------------------
__global__ __launch_bounds__(128) void spmm_scatter(
    const float* __restrict__ H, const int* __restrict__ src,
    const int* __restrict__ dst, const float* __restrict__ ew,
    float* __restrict__ Sv)
{
  const int e = blockIdx.x;
  const int s = src[e], d = dst[e];
  const float wt = ew[e];
  const float* hp = H + (size_t)s * HID;
  float* sp = Sv + (size_t)d * HID;
  const int c = threadIdx.x << 2;
#pragma unroll
  for (int j = 0; j < 4; ++j)
    atomicAdd(&sp[c + j], hp[c + j] * wt);
}

__global__ void zero_f32(float* __restrict__ p, size_t n) {
  size_t i = (size_t)blockIdx.x * blockDim.x + threadIdx.x;
  if (i < n) p[i] = 0.0f;
}

__global__ void layer_epilogue(float* __restrict__ x, const float* __restrict__ Sv,
                               const float* __restrict__ b, int do_relu)
{
  size_t i = (size_t)blockIdx.x * blockDim.x + threadIdx.x;
  if (i >= (size_t)N_NODES * HID) return;
  float v = Sv[i] + b[i & (HID - 1)];
  x[i] = do_relu ? fmaxf(v, 0.0f) : (x[i] + v);
}

__global__ __launch_bounds__(256) void recon_kernel(
    const float* __restrict__ R, const float* __restrict__ X0,
    float* __restrict__ accum)
{
  __shared__ float s1[256], s2[256], s3[256];
  const int row = blockIdx.x, tid = threadIdx.x;
  float a2 = 0.f, b2 = 0.f, ab = 0.f;
  for (int c = tid; c < FT0; c += 256) {
    float a = R[(size_t)row * FT0 + c];
    float b = X0[(size_t)row * FT0 + c];
    a2 += a * a; b2 += b * b; ab += a * b;
  }
  s1[tid] = a2; s2[tid] = b2; s3[tid] = ab;
  __syncthreads();
  for (int s = 128; s > 0; s >>= 1) {
    if (tid < s) { s1[tid] += s1[tid + s]; s2[tid] += s2[tid + s]; s3[tid] += s3[tid + s]; }
    __syncthreads();
  }
  if (tid == 0) {
    float na = fmaxf(sqrtf(s1[0]), 1e-12f);
    float nb = fmaxf(sqrtf(s2[0]), 1e-12f);
    float t  = 1.0f - s3[0] / (na * nb);
    atomicAdd(accum, t * t * t / (float)SZ0);
  }
}

__global__ void init_accum(float* a) { if (threadIdx.x < 2) a[threadIdx.x] = 0.0f; }

__global__ void finalize_losses(const float* __restrict__ acc, float* __restrict__ out2) {
  out2[0] = acc[0] / ((float)N_NODES * (float)N_NODES);
  out2[1] = acc[1];
}

// ---------------------------------------------------------------------------
extern "C" void kernel_launch(void* const* d_in, const int* in_sizes, int n_in,
                              void* d_out, int out_size, void* d_ws, size_t ws_size,
                              hipStream_t stream) {
  (void)in_sizes; (void)n_in; (void)out_size; (void)ws_size;
  const float* X0     = (const float*)d_in[0];
  const float* X1     = (const float*)d_in[1];
  const float* X2     = (const float*)d_in[2];
  const int*   src    = (const int*)d_in[3];
  const int*   dst    = (const int*)d_in[4];
  const float* edge_w = (const float*)d_in[5];
  const float* adj    = (const float*)d_in[6];
  const float* fcW0 = (const float*)d_in[7],  *fcb0 = (const float*)d_in[8];
  const float* fcW1 = (const float*)d_in[9],  *fcb1 = (const float*)d_in[10];
  const float* fcW2 = (const float*)d_in[11], *fcb2 = (const float*)d_in[12];
  const float* encW = (const float*)d_in[13], *encB = (const float*)d_in[14];
  const float* decW = (const float*)d_in[15], *decB = (const float*)d_in[16];
  const float* outW0 = (const float*)d_in[17], *outb0 = (const float*)d_in[18];
  const float* outW1 = (const float*)d_in[19], *outb1 = (const float*)d_in[20];
  const float* outW2 = (const float*)d_in[21], *outb2 = (const float*)d_in[22];
  float* out = (float*)d_out;

  const size_t xh = (size_t)N_NODES * HID;
  float* x   = (float*)d_ws;
  float* h   = x + xh;
  float* Sv  = h + xh;
  float* acc = Sv + xh;

  const size_t off_r1  = (size_t)SZ0 * FT0;
  const size_t off_r2  = off_r1 + (size_t)SZ1 * FT1;
  const size_t off_emb = off_r2 + (size_t)SZ2 * FT2;

  auto cdiv = [](int a, int b) { return (a + b - 1) / b; };
  dim3 blk(256);
  const int zgrid = (int)((xh + 255) / 256);

  init_accum<<<1, 32, 0, stream>>>(acc);

  gemm_wmma_f32<1><<<dim3(cdiv(HID, 64), cdiv(SZ0, 128)), blk, 0, stream>>>(
      X0, fcW0, fcb0, x, SZ0, FT0, HID);
  gemm_wmma_f32<1><<<dim3(cdiv(HID, 64), cdiv(SZ1, 128)), blk, 0, stream>>>(
      X1, fcW1, fcb1, x + (size_t)SZ0 * HID, SZ1, FT1, HID);
  gemm_wmma_f32<1><<<dim3(cdiv(HID, 64), cdiv(SZ2, 128)), blk, 0, stream>>>(
      X2, fcW2, fcb2, x + (size_t)(SZ0 + SZ1) * HID, SZ2, FT2, HID);

  for (int st = 0; st < 2; ++st) {
    const float* Wl = st ? decW : encW;
    const float* bl = st ? decB : encB;
    for (int i = 0; i < NLAYER; ++i) {
      gemm_wmma_f32<0><<<dim3(cdiv(HID, 64), cdiv(N_NODES, 128)), blk, 0, stream>>>(
          x, Wl + (size_t)i * HID * HID, nullptr, h, N_NODES, HID, HID);
      zero_f32<<<zgrid, 256, 0, stream>>>(Sv, xh);
      spmm_scatter<<<NEDGE, 128, 0, stream>>>(h, src, dst, edge_w, Sv);
      layer_epilogue<<<zgrid, 256, 0, stream>>>(x, Sv, bl + (size_t)i * HID,
                                                (i == 0) ? 1 : 0);
    }
  }

  gemm_wmma_f32<1><<<dim3(cdiv(FT0, 64), cdiv(SZ0, 128)), blk, 0, stream>>>(
      x, outW0, outb0, out, SZ0, HID, FT0);
  gemm_wmma_f32<1><<<dim3(cdiv(FT1, 64), cdiv(SZ1, 128)), blk, 0, stream>>>(
      x + (size_t)SZ0 * HID, outW1, outb1, out + off_r1, SZ1, HID, FT1);
  gemm_wmma_f32<1><<<dim3(cdiv(FT2, 64), cdiv(SZ2, 128)), blk, 0, stream>>>(
      x + (size_t)(SZ0 + SZ1) * HID, outW2, outb2, out + off_r2, SZ2, HID, FT2);

  zloss_wmma<<<dim3(cdiv(N_NODES, 64), cdiv(N_NODES, 128)), blk, 0, stream>>>(
      x, adj, acc);

  recon_kernel<<<SZ0, 256, 0, stream>>>(out, X0, acc + 1);

  finalize_losses<<<1, 1, 0, stream>>>(acc, out + off_emb);
}